// SimpleMoELayer_59657095741908
// MI455X (gfx1250) — compile-verified
//
#include <hip/hip_runtime.h>
#include <hip/hip_bf16.h>

#define T_TOK 8192
#define DM    1024
#define FF    4096
#define NE    8

typedef float v2f __attribute__((ext_vector_type(2)));
typedef float v8f __attribute__((ext_vector_type(8)));

// Branch-free tanh-approx GELU (matches jax.nn.gelu approximate=True).
// exp argument is always <= 0 so __expf (v_exp_f32) is safe: no libm branches.
__device__ __forceinline__ float gelu_tanh(float x) {
    const float k0 = 0.7978845608028654f;   // sqrt(2/pi)
    const float k1 = 0.044715f;
    float u  = k0 * (x + k1 * x * x * x);
    float au = fabsf(u);
    float t  = __expf(-2.0f * au);          // in (0,1], underflows to 0 for big |u|
    float th = (1.0f - t) / (1.0f + t);     // tanh(|u|)
    th = copysignf(th, u);
    return 0.5f * x * (1.0f + th);
}

// ---------------- workspace layout (32-bit words) ----------------
// [0..7]    int   cnt[NE]          routing list fill counters
// [8..15]   float top1cnt[NE]      #tokens whose top-1 == e
// [16..23]  float probsum[NE]      sum over tokens of softmax prob
// [32 ..]                 int   tok_list[NE][T_TOK]
// [32+NE*T_TOK ..]        float w_list [NE][T_TOK]

// Zero output accumulator + counters. Grid covers exactly T*DM/4 float4 stores.
__global__ void moe_init(float* __restrict__ out, unsigned int* __restrict__ ws_u32) {
    long long tid = (long long)blockIdx.x * blockDim.x + threadIdx.x;
    float4 z = make_float4(0.f, 0.f, 0.f, 0.f);
    *(float4*)(out + tid * 4) = z;
    if (tid < 32) ws_u32[tid] = 0u;
}

// Gating: logits -> softmax -> top2 -> routing lists + aux-loss accumulators.
__global__ __launch_bounds__(256) void moe_gate(const float* __restrict__ x,
                                                const float* __restrict__ Wg,
                                                const float* __restrict__ bg,
                                                int*   __restrict__ cnt,
                                                float* __restrict__ top1cnt,
                                                float* __restrict__ probsum,
                                                int*   __restrict__ tok_list,
                                                float* __restrict__ w_list) {
    __shared__ float wg_s[DM * NE];   // 32 KB, broadcast reads
    __shared__ float blk_ps[NE];
    __shared__ float blk_c[NE];
    const int tid = threadIdx.x;
    for (int i = tid; i < DM * NE; i += 256) wg_s[i] = Wg[i];
    if (tid < NE) { blk_ps[tid] = 0.f; blk_c[tid] = 0.f; }
    __syncthreads();

    const int t = blockIdx.x * 256 + tid;
    float acc[NE];
    #pragma unroll
    for (int e = 0; e < NE; e++) acc[e] = bg[e];

    const float* xr = x + (long long)t * DM;
    for (int d = 0; d < DM; d += 4) {
        float4 xv = *(const float4*)(xr + d);
        #pragma unroll
        for (int e = 0; e < NE; e++) {
            acc[e] += xv.x * wg_s[(d + 0) * NE + e] + xv.y * wg_s[(d + 1) * NE + e]
                    + xv.z * wg_s[(d + 2) * NE + e] + xv.w * wg_s[(d + 3) * NE + e];
        }
    }
    // softmax over 8
    float mx = acc[0];
    #pragma unroll
    for (int e = 1; e < NE; e++) mx = fmaxf(mx, acc[e]);
    float p[NE], s = 0.f;
    #pragma unroll
    for (int e = 0; e < NE; e++) { p[e] = __expf(acc[e] - mx); s += p[e]; }
    float inv = 1.f / s;
    #pragma unroll
    for (int e = 0; e < NE; e++) p[e] *= inv;
    // top-2 (ties -> lowest index, matching lax.top_k)
    int i1 = 0; float p1 = p[0];
    #pragma unroll
    for (int e = 1; e < NE; e++) if (p[e] > p1) { p1 = p[e]; i1 = e; }
    int i2 = (i1 == 0) ? 1 : 0; float p2 = p[i2];
    #pragma unroll
    for (int e = 0; e < NE; e++) if (e != i1 && p[e] > p2) { p2 = p[e]; i2 = e; }

    int pos1 = atomicAdd(&cnt[i1], 1);
    tok_list[i1 * T_TOK + pos1] = t;  w_list[i1 * T_TOK + pos1] = p1;
    int pos2 = atomicAdd(&cnt[i2], 1);
    tok_list[i2 * T_TOK + pos2] = t;  w_list[i2 * T_TOK + pos2] = p2;

    atomicAdd(&blk_c[i1], 1.0f);
    #pragma unroll
    for (int e = 0; e < NE; e++) atomicAdd(&blk_ps[e], p[e]);
    __syncthreads();
    if (tid < NE) {
        atomicAdd(&top1cnt[tid], blk_c[tid]);
        atomicAdd(&probsum[tid], blk_ps[tid]);
    }
}

// Expert FFN: per block = one expert e, one 16-token tile.
// 8 waves: h-phase -> each wave one 16x16 h tile per 128-wide F chunk (K=1024),
// y-phase -> 8 persistent 16x16 accumulators per wave (N=1024 split across waves).
__global__ __launch_bounds__(256) void moe_ffn(const float* __restrict__ x,
                                               const float* __restrict__ W1,
                                               const float* __restrict__ b1,
                                               const float* __restrict__ W2,
                                               const float* __restrict__ b2,
                                               float* __restrict__ out,
                                               const int*   __restrict__ cnt,
                                               const int*   __restrict__ tok_list,
                                               const float* __restrict__ w_list) {
    const int e  = blockIdx.y;
    const int nt = cnt[e];
    const int t0 = blockIdx.x * 16;
    if (t0 >= nt) return;                       // block-uniform exit

    __shared__ float xs[16][DM + 4];            // +4 pad: A-frag reads hit distinct banks
    __shared__ float hs[16][128 + 4];
    __shared__ int   stok[16];
    __shared__ float sw[16];

    const int tid  = threadIdx.x;
    const int lane = tid & 31;
    const int wv   = tid >> 5;                  // wave id 0..7

    if (tid < 16) {
        int idx = t0 + tid;
        if (idx < nt) { stok[tid] = tok_list[e * T_TOK + idx]; sw[tid] = w_list[e * T_TOK + idx]; }
        else          { stok[tid] = -1;                        sw[tid] = 0.f; }
    }
    __syncthreads();
    // gather 16 token rows of x into LDS (zeros for pad rows)
    for (int i = tid; i < 16 * (DM / 4); i += 256) {
        int row = i >> 8;                       // DM/4 == 256
        int c   = i & 255;
        float4 v = make_float4(0.f, 0.f, 0.f, 0.f);
        int tk = stok[row];
        if (tk >= 0) v = *(const float4*)(x + (long long)tk * DM + 4 * c);
        *(float4*)(&xs[row][4 * c]) = v;
    }
    __syncthreads();

    const int m    = lane & 15;                 // A rows / B-N col / C-N col
    const int koff = (lane < 16) ? 0 : 2;       // K slice held by this half-wave
    const int rowh = 8 * (lane >> 4);           // C/D: vgpr r holds rows r and r+8

    const float* W1e = W1 + (long long)e * DM * FF;
    const float* W2e = W2 + (long long)e * FF * DM;

    v8f yacc[8];
    #pragma unroll
    for (int a = 0; a < 8; a++) yacc[a] = (v8f){0.f,0.f,0.f,0.f,0.f,0.f,0.f,0.f};

    const int fwv = 16 * wv;                    // this wave's h-column slab within chunk
    const float* b1e = b1 + e * FF;

    for (int f0 = 0; f0 < FF; f0 += 128) {
        // ---- h = gelu(x @ W1 + b1), 16x16 per wave, K = 1024 ----
        v8f hacc = (v8f){0.f,0.f,0.f,0.f,0.f,0.f,0.f,0.f};
        const int fcol = f0 + fwv + m;
        const float* w1p = W1e + fcol;          // element [d][fcol] at w1p[d*FF]
        float bias1 = b1e[fcol];                // load early, overlaps with K loop
        for (int k = 0; k < DM; k += 4) {
            v2f a; a.x = xs[m][k + koff];       a.y = xs[m][k + koff + 1];
            v2f b; b.x = w1p[(long long)(k + koff) * FF];
                   b.y = w1p[(long long)(k + koff + 1) * FF];
            hacc = __builtin_amdgcn_wmma_f32_16x16x4_f32(false, a, false, b,
                                                         (short)0, hacc, false, false);
        }
        #pragma unroll
        for (int r = 0; r < 8; r++)
            hs[r + rowh][fwv + m] = gelu_tanh(hacc[r] + bias1);
        __syncthreads();

        // ---- y += h_chunk @ W2[f0:f0+128, :], 8 accumulators per wave ----
        for (int kc = 0; kc < 128; kc += 4) {
            v2f a; a.x = hs[m][kc + koff];      a.y = hs[m][kc + koff + 1];
            const float* w2p = W2e + (long long)(f0 + kc + koff) * DM;
            #pragma unroll
            for (int ai = 0; ai < 8; ai++) {
                int ncol = 128 * wv + 16 * ai + m;
                v2f b; b.x = w2p[ncol]; b.y = w2p[DM + ncol];
                yacc[ai] = __builtin_amdgcn_wmma_f32_16x16x4_f32(false, a, false, b,
                                                                 (short)0, yacc[ai], false, false);
            }
        }
        __syncthreads();
    }

    // ---- epilogue: out[t] += w * (y + b2) ----
    const float* b2e = b2 + e * DM;
    #pragma unroll
    for (int ai = 0; ai < 8; ai++) {
        int ncol = 128 * wv + 16 * ai + m;
        float bias2 = b2e[ncol];
        #pragma unroll
        for (int r = 0; r < 8; r++) {
            int mm = r + rowh;
            int tk = stok[mm];
            if (tk >= 0) {
                float v = (yacc[ai][r] + bias2) * sw[mm];
                unsafeAtomicAdd(out + (long long)tk * DM + ncol, v);
            }
        }
    }
}

__global__ void moe_aux(const float* __restrict__ top1cnt,
                        const float* __restrict__ probsum,
                        float* __restrict__ out_aux) {
    if (threadIdx.x == 0) {
        float aux = 0.f;
        for (int e = 0; e < NE; e++) {
            float frac = top1cnt[e] / ((float)T_TOK + 1e-8f);
            float avgp = probsum[e] / (float)T_TOK;
            aux += frac * avgp;
        }
        *out_aux = aux * (float)NE;
    }
}

extern "C" void kernel_launch(void* const* d_in, const int* in_sizes, int n_in,
                              void* d_out, int out_size, void* d_ws, size_t ws_size,
                              hipStream_t stream) {
    const float* x  = (const float*)d_in[0];
    const float* Wg = (const float*)d_in[1];
    const float* bg = (const float*)d_in[2];
    const float* W1 = (const float*)d_in[3];
    const float* b1 = (const float*)d_in[4];
    const float* W2 = (const float*)d_in[5];
    const float* b2 = (const float*)d_in[6];
    float* out = (float*)d_out;

    unsigned int* ws32 = (unsigned int*)d_ws;
    int*   cnt      = (int*)ws32;
    float* top1cnt  = (float*)(ws32 + 8);
    float* probsum  = (float*)(ws32 + 16);
    int*   tok_list = (int*)(ws32 + 32);
    float* w_list   = (float*)(ws32 + 32 + NE * T_TOK);

    const long long outElems = (long long)T_TOK * DM;   // 8,388,608 (div by 1024)
    moe_init<<<(int)(outElems / (256 * 4)), 256, 0, stream>>>(out, ws32);
    moe_gate<<<T_TOK / 256, 256, 0, stream>>>(x, Wg, bg, cnt, top1cnt, probsum,
                                              tok_list, w_list);
    moe_ffn<<<dim3(T_TOK / 16, NE), 256, 0, stream>>>(x, W1, b1, W2, b2, out,
                                                      cnt, tok_list, w_list);
    moe_aux<<<1, 32, 0, stream>>>(top1cnt, probsum, out + outElems);
}